// BlockSparseFactIndex_16346645528845
// MI455X (gfx1250) — compile-verified
//
#include <hip/hip_runtime.h>
#include <stdint.h>

#define PP     50
#define EE     2000
#define MCAND  64
#define PE     (PP * EE)   /* 100000 buckets per CSR */

typedef __attribute__((ext_vector_type(4))) float v4f;

// ---------------------------------------------------------------------------
// Kernel 1: bucket histograms for both CSRs (ps: key=fp*E+fs, po: key=fp*E+fo)
// ---------------------------------------------------------------------------
__global__ void count_kernel(const int* __restrict__ facts, int F,
                             int* __restrict__ ps_cnt, int* __restrict__ po_cnt) {
    int i = blockIdx.x * blockDim.x + threadIdx.x;
    if (i >= F) return;
    int fp = __builtin_nontemporal_load(&facts[3 * i + 0]);
    int fs = __builtin_nontemporal_load(&facts[3 * i + 1]);
    int fo = __builtin_nontemporal_load(&facts[3 * i + 2]);
    atomicAdd(&ps_cnt[fp * EE + fs], 1);
    atomicAdd(&po_cnt[fp * EE + fo], 1);
}

// ---------------------------------------------------------------------------
// Kernel 2: exclusive scan of PE counts -> offsets[0..PE] and cursor init.
// Single workgroup (1024 threads = 32 waves). Chunks are streamed into LDS
// with double-buffered gfx1250 async copies (global_load_async_to_lds_b32 +
// s_wait_asynccnt): chunk c+1 is in flight while chunk c is scanned.
// The async issue is UNCONDITIONAL per wave (OOB lanes clamp their source
// index, consume side masks them to 0) so ASYNCcnt is uniform across waves
// and `s_wait_asynccnt 0x1` provably releases the chunk being consumed
// (async loads complete in order). `cur` may alias `cnt`.
// ---------------------------------------------------------------------------
__device__ __forceinline__ void async_load_dword(unsigned lds_byte_off,
                                                 unsigned glb_byte_off,
                                                 const int* base) {
    asm volatile("global_load_async_to_lds_b32 %0, %1, %2"
                 :: "v"(lds_byte_off), "v"(glb_byte_off), "s"(base)
                 : "memory");
}

__global__ void scan_kernel(const int* cnt, int n, int* off, int* cur) {
    __shared__ int sh[1024];
    __shared__ int abuf[2][1024];
    __shared__ int carry_sh;
    const int tid = threadIdx.x;
    if (tid == 0) carry_sh = 0;

    const int nc = (n + 1023) >> 10;   // number of 1024-chunks

    // Prologue: issue chunk 0 into abuf[0] (clamped, unconditional).
    {
        int i0 = tid < n ? tid : (n - 1);
        async_load_dword((unsigned)(uintptr_t)&abuf[0][tid],
                         (unsigned)(i0 * 4), cnt);
    }

    int p = 0;
    for (int c = 0; c < nc; ++c) {
        const int base = c << 10;
        const bool has_next = (c + 1) < nc;
        if (has_next) {
            int i2 = base + 1024 + tid;
            if (i2 >= n) i2 = n - 1;                    // clamp, keep issue uniform
            async_load_dword((unsigned)(uintptr_t)&abuf[1 - p][tid],
                             (unsigned)(i2 * 4), cnt);
            asm volatile("s_wait_asynccnt 0x1" ::: "memory");  // chunk c landed
        } else {
            asm volatile("s_wait_asynccnt 0x0" ::: "memory");
        }
        __syncthreads();   // async LDS writes visible workgroup-wide

        int i = base + tid;
        int v = (i < n) ? abuf[p][tid] : 0;
        sh[tid] = v;
        __syncthreads();
        #pragma unroll
        for (int d = 1; d < 1024; d <<= 1) {
            int t = (tid >= d) ? sh[tid - d] : 0;
            __syncthreads();
            sh[tid] += t;
            __syncthreads();
        }
        int incl  = sh[tid];
        int carry = carry_sh;            // value from previous chunk
        if (i < n) {
            int excl = carry + incl - v;
            off[i] = excl;
            cur[i] = excl;
        }
        __syncthreads();                 // all reads of carry_sh done
        if (tid == 1023) carry_sh = carry + incl;
        __syncthreads();                 // also guards abuf[p] reuse next iter
        p = 1 - p;
    }
    if (tid == 0) off[n] = carry_sh;
}

// ---------------------------------------------------------------------------
// Kernel 3: scatter values into CSR slots (intra-bucket order arbitrary here;
// fixed up by the per-bucket sort below, making the result deterministic).
// ---------------------------------------------------------------------------
__global__ void scatter_kernel(const int* __restrict__ facts, int F,
                               int* __restrict__ ps_cur, int* __restrict__ po_cur,
                               int* __restrict__ ps_vals, int* __restrict__ po_vals) {
    int i = blockIdx.x * blockDim.x + threadIdx.x;
    if (i >= F) return;
    int fp = __builtin_nontemporal_load(&facts[3 * i + 0]);
    int fs = __builtin_nontemporal_load(&facts[3 * i + 1]);
    int fo = __builtin_nontemporal_load(&facts[3 * i + 2]);
    int p1 = atomicAdd(&ps_cur[fp * EE + fs], 1);
    ps_vals[p1] = fo;
    int p2 = atomicAdd(&po_cur[fp * EE + fo], 1);
    po_vals[p2] = fs;
}

// ---------------------------------------------------------------------------
// Kernel 4: sort each bucket ascending, both CSRs in one grid. One thread per
// bucket (Poisson(20) sizes). Buckets are contiguous + L2-resident; prefetch
// the range (global_prefetch_b8) ahead of the latency-chained insertion sort.
// ---------------------------------------------------------------------------
__global__ void bucket_sort_kernel(const int* __restrict__ ps_off,
                                   int* __restrict__ ps_vals,
                                   const int* __restrict__ po_off,
                                   int* __restrict__ po_vals, int nb_each) {
    int g = blockIdx.x * blockDim.x + threadIdx.x;
    if (g >= 2 * nb_each) return;
    const int* off  = (g < nb_each) ? ps_off  : po_off;
    int*       vals = (g < nb_each) ? ps_vals : po_vals;
    int b = (g < nb_each) ? g : g - nb_each;

    int s = off[b], e = off[b + 1];
    if (e - s > 1) {
        __builtin_prefetch(&vals[s], 1, 3);
        for (int i = s + 1; i < e; ++i) {
            int key = vals[i];
            int j = i - 1;
            while (j >= s && vals[j] > key) { vals[j + 1] = vals[j]; --j; }
            vals[j + 1] = key;
        }
    }
}

// ---------------------------------------------------------------------------
// Kernel 5: queries. 16 threads per query row, 4 candidates each.
// Must reproduce the reference's clamped gather even at invalid positions.
// Outputs are write-once streaming -> non-temporal 128-bit stores.
// ---------------------------------------------------------------------------
__global__ void query_kernel(const int* __restrict__ preds,
                             const int* __restrict__ bound,
                             const int* __restrict__ dir,
                             int N, int F,
                             const int* __restrict__ ps_off, const int* __restrict__ po_off,
                             const int* __restrict__ ps_vals, const int* __restrict__ po_vals,
                             float* __restrict__ out_cand, float* __restrict__ out_valid) {
    long long t = (long long)blockIdx.x * blockDim.x + threadIdx.x;
    if (t >= (long long)N * (MCAND / 4)) return;
    int n    = (int)(t >> 4);
    int pos0 = ((int)t & 15) * 4;

    int  key    = preds[n] * EE + bound[n];
    bool is_obj = (dir[n] == 0);
    const int* __restrict__ off  = is_obj ? ps_off  : po_off;
    const int* __restrict__ vals = is_obj ? ps_vals : po_vals;

    int start = off[key];
    int cnt   = off[key + 1] - start;
    cnt = cnt < 0 ? 0 : (cnt > MCAND ? MCAND : cnt);

    v4f c, v;
    #pragma unroll
    for (int k = 0; k < 4; ++k) {
        int pos = pos0 + k;
        int gi  = start + pos;
        gi = gi < 0 ? 0 : gi;
        gi = gi >= F ? F - 1 : gi;
        c[k] = (float)vals[gi];
        v[k] = (pos < cnt) ? 1.0f : 0.0f;
    }
    size_t o = (size_t)n * MCAND + (size_t)pos0;
    __builtin_nontemporal_store(c, (v4f*)(out_cand  + o));
    __builtin_nontemporal_store(v, (v4f*)(out_valid + o));
}

// ---------------------------------------------------------------------------
extern "C" void kernel_launch(void* const* d_in, const int* in_sizes, int n_in,
                              void* d_out, int out_size, void* d_ws, size_t ws_size,
                              hipStream_t stream) {
    const int* facts = (const int*)d_in[0];   // (F,3) row-major: fp, fs, fo
    const int* preds = (const int*)d_in[1];
    const int* bound = (const int*)d_in[2];
    const int* dir   = (const int*)d_in[3];

    const int F = in_sizes[0] / 3;
    const int N = in_sizes[1];

    // Workspace layout (ints): 2F + 2(PE+1) + 2PE  ~= 17.6 MB
    int* ws      = (int*)d_ws;
    int* ps_vals = ws;  ws += F;
    int* po_vals = ws;  ws += F;
    int* ps_off  = ws;  ws += PE + 1;
    int* po_off  = ws;  ws += PE + 1;
    int* ps_cnt  = ws;  ws += PE;      // becomes scatter cursor after scan
    int* po_cnt  = ws;  ws += PE;

    (void)hipMemsetAsync(ps_cnt, 0, sizeof(int) * PE, stream);
    (void)hipMemsetAsync(po_cnt, 0, sizeof(int) * PE, stream);

    const int TB = 256;
    count_kernel<<<(F + TB - 1) / TB, TB, 0, stream>>>(facts, F, ps_cnt, po_cnt);

    scan_kernel<<<1, 1024, 0, stream>>>(ps_cnt, PE, ps_off, ps_cnt);
    scan_kernel<<<1, 1024, 0, stream>>>(po_cnt, PE, po_off, po_cnt);

    scatter_kernel<<<(F + TB - 1) / TB, TB, 0, stream>>>(facts, F, ps_cnt, po_cnt,
                                                         ps_vals, po_vals);

    bucket_sort_kernel<<<(2 * PE + TB - 1) / TB, TB, 0, stream>>>(ps_off, ps_vals,
                                                                  po_off, po_vals, PE);

    long long qthreads = (long long)N * (MCAND / 4);
    int qblocks = (int)((qthreads + TB - 1) / TB);
    float* out_cand  = (float*)d_out;
    float* out_valid = out_cand + (size_t)N * MCAND;
    query_kernel<<<qblocks, TB, 0, stream>>>(preds, bound, dir, N, F,
                                             ps_off, po_off, ps_vals, po_vals,
                                             out_cand, out_valid);
}